// RelativeTemEmbedding_65773129170979
// MI455X (gfx1250) — compile-verified
//
#include <hip/hip_runtime.h>
#include <hip/hip_bf16.h>

// Problem constants (from reference)
#define Bv 4
#define Hv 16
#define Lv 2048
#define Dv 64
#define NB 19          // 2*10-1 buckets
#define BH (Bv*Hv)     // 64

typedef __attribute__((ext_vector_type(2))) float v2f;
typedef __attribute__((ext_vector_type(8))) float v8f;
typedef __attribute__((ext_vector_type(4))) float f4;

// ---------------------------------------------------------------------------
// Kernel 1: lut[b,h,l,n] = sum_d Q[b,h,l,d] * W[h,d,n]   (fp32 WMMA 16x16x4)
// One wave handles a 16(l) x 19(n) tile; 8 waves per block, all sharing the
// same head h (128 l-tiles per (b,h), 8 tiles/block => no boundary crossing),
// so W[h] is staged once per block into LDS, zero-padded 19 -> 32 columns.
// This makes every B-fragment read an unconditional conflict-free ds_load
// (no exec-mask predication branches in the k-loop).
// grid.x = B*H*(L/16)/8 = 1024, block = 256
// ---------------------------------------------------------------------------
__global__ __launch_bounds__(256) void proj_wmma(
    const float* __restrict__ Q,   // [B,H,L,D]
    const float* __restrict__ W,   // [H,D,NB]
    float* __restrict__ lut)       // [B,H,L,NB] (workspace)
{
    __shared__ float s_w[Dv][32];                  // 8 KB, cols 19..31 = 0

    const int tid   = threadIdx.x;
    const int lane  = tid & 31;
    const int wave  = tid >> 5;
    const int tile0 = blockIdx.x * 8;
    const int bh    = tile0 >> 7;                  // same for whole block
    const int h     = bh & (Hv - 1);

    // stage W[h] padded to 32 columns (2048 floats, 8 per thread)
    const float* __restrict__ wsrc = W + (size_t)h * Dv * NB;
    for (int i = tid; i < Dv * 32; i += 256) {
        const int r = i >> 5, c = i & 31;
        s_w[r][c] = (c < NB) ? wsrc[r * NB + c] : 0.0f;
    }
    __syncthreads();

    const int tile  = tile0 + wave;
    const int ltile = tile & (Lv / 16 - 1);
    const int l0    = ltile * 16;
    const float* __restrict__ q = Q + ((size_t)bh * Lv + l0) * Dv;  // 16 x 64

    // A-matrix 16x4 f32 layout (ISA 7.12.2): lanes 0-15 -> M=lane, v0=K0 v1=K1;
    // lanes 16-31 -> M=lane-16, v0=K2 v1=K3.
    const int m  = lane & 15;
    const int kq = (lane >> 4) * 2;     // 0 or 2
    // B-matrix 4x16 f32: v0: K=0 (lanes 0-15), K=1 (lanes 16-31); v1: K=2, K=3.
    const int kb = lane >> 4;           // 0 or 1
    const int n  = lane & 15;

    v8f c0 = {};   // columns 0..15
    v8f c1 = {};   // columns 16..18 (16..31 padded with zeros)

    #pragma unroll
    for (int kk = 0; kk < 16; ++kk) {
        const int kbase = kk * 4;
        v2f a;
        a.x = q[m * Dv + kbase + kq];
        a.y = q[m * Dv + kbase + kq + 1];

        v2f b0, b1;
        b0.x = s_w[kbase + kb    ][n];
        b0.y = s_w[kbase + kb + 2][n];
        b1.x = s_w[kbase + kb    ][16 + n];
        b1.y = s_w[kbase + kb + 2][16 + n];

        c0 = __builtin_amdgcn_wmma_f32_16x16x4_f32(false, a, false, b0,
                                                   (short)0, c0, false, false);
        c1 = __builtin_amdgcn_wmma_f32_16x16x4_f32(false, a, false, b1,
                                                   (short)0, c1, false, false);
    }

    // C/D 16x16 f32 layout: VGPR r -> M = r + (lane>=16 ? 8 : 0), N = lane&15.
    float* __restrict__ outrow = lut + ((size_t)bh * Lv + l0) * NB;
    const int  mbase    = (lane >> 4) * 8;
    const bool hi_valid = (16 + n) < NB;
    #pragma unroll
    for (int r = 0; r < 8; ++r) {
        const int M = r + mbase;
        outrow[M * NB + n] = c0[r];
        if (hi_valid) outrow[M * NB + 16 + n] = c1[r];
    }
}

// ---------------------------------------------------------------------------
// Kernel 2 (bandwidth-bound): out[b,h,l,j] = lut[b,h,l, t[l]-t[j]+9]
// One block per l. Bucket indices cached once as bytes in LDS; all 64 lut
// rows for this l staged in LDS; 8 consecutive j per thread per row,
// written with two non-temporal b128 stores (output >> L2, keep t/lut hot).
// grid.x = L = 2048, block = 256
// ---------------------------------------------------------------------------
__global__ __launch_bounds__(256) void gather_bcast(
    const int*   __restrict__ t,     // [B,L] -> uses row 0 only
    const float* __restrict__ lut,   // [B,H,L,NB]
    float*       __restrict__ out)   // [B,H,L,L]
{
    __shared__ __align__(8) unsigned char s_idx[Lv];   // 2 KB
    __shared__ float s_row[BH][20];                    // 5 KB, 19->20 pad

    const int l   = blockIdx.x;
    const int tid = threadIdx.x;
    const int tl  = t[l] + NB / 2;

    // bucket indices for this query row (values in [0,19))
    for (int j = tid; j < Lv; j += 256)
        s_idx[j] = (unsigned char)(tl - t[j]);

    // all 64 (b,h) logit rows for this l
    for (int i = tid; i < BH * NB; i += 256) {
        const int bh = i / NB;
        const int nn = i - bh * NB;
        s_row[bh][nn] = lut[((size_t)bh * Lv + l) * NB + nn];
    }
    __syncthreads();

    const int j0 = tid * 8;
    const unsigned long long pk = *(const unsigned long long*)(s_idx + j0);
    int idx[8];
    #pragma unroll
    for (int i = 0; i < 8; ++i)
        idx[i] = (int)((pk >> (8 * i)) & 0xFF);

    for (int bh = 0; bh < BH; ++bh) {
        const float* __restrict__ row = s_row[bh];
        f4 o0, o1;
        o0.x = row[idx[0]]; o0.y = row[idx[1]];
        o0.z = row[idx[2]]; o0.w = row[idx[3]];
        o1.x = row[idx[4]]; o1.y = row[idx[5]];
        o1.z = row[idx[6]]; o1.w = row[idx[7]];
        float* p = out + ((size_t)bh * Lv + l) * Lv + j0;
        __builtin_nontemporal_store(o0, (f4*)p);
        __builtin_nontemporal_store(o1, (f4*)(p + 4));
    }
}

// ---------------------------------------------------------------------------
extern "C" void kernel_launch(void* const* d_in, const int* in_sizes, int n_in,
                              void* d_out, int out_size, void* d_ws, size_t ws_size,
                              hipStream_t stream) {
    const int*   t = (const int*)d_in[0];    // temporal_tokens [B,L] int32
    const float* Q = (const float*)d_in[1];  // query_emb [B,H,L,D] f32
    const float* W = (const float*)d_in[2];  // lookup_table_weight [H,D,NB] f32
    float* out = (float*)d_out;              // [B,H,L,L] f32
    float* lut = (float*)d_ws;               // [B,H,L,NB] f32 scratch (~10 MB)

    // 8192 wave-tiles of 16 rows, 8 waves (256 threads) per block
    proj_wmma<<<dim3((Bv * Hv * (Lv / 16)) / 8), dim3(256), 0, stream>>>(Q, W, lut);
    // one block per query position l
    gather_bcast<<<dim3(Lv), dim3(256), 0, stream>>>(t, lut, out);
}